// Loss_34608846471397
// MI455X (gfx1250) — compile-verified
//
#include <hip/hip_runtime.h>
#include <math.h>

// ---------------------------------------------------------------------------
// Center-loss kernel for MI455X (gfx1250, wave32).
//   intra = mean_i clip(||f_i - center[label_i]||, 1e-12, 1e12)
//   inter = (2 / ||cen[998]-cen[999]||) / (1000*999)
//       cen[c] = (center[c] + sum_{label==c} f) / max(count_c, 1)
// Workspace layout (floats):
//   ws[0]          : intra distance accumulator
//   ws[1], ws[2]   : counts for class 998, 999
//   ws[16 .. 16+1023] : feature sums for class 998 (512) then 999 (512)
// ---------------------------------------------------------------------------

typedef float v2f __attribute__((ext_vector_type(2)));
typedef float v8f __attribute__((ext_vector_type(8)));

#define FD   512
#define NB   32768
#define WS_SUMS 16

__global__ void zero_ws_kernel(float* ws, int n) {
    int i = blockIdx.x * blockDim.x + threadIdx.x;
    if (i < n) ws[i] = 0.0f;
}

// One wave (32 lanes) handles 16 rows. Block = 256 threads = 8 waves = 128 rows.
// Grid = 32768 / 128 = 256 blocks.
__global__ void __launch_bounds__(256)
center_loss_main_kernel(const float* __restrict__ feat,
                        const int*   __restrict__ labels,
                        const float* __restrict__ center,
                        float* ws) {
    const int lane    = threadIdx.x & 31;
    const int wave    = threadIdx.x >> 5;
    const int rowbase = (blockIdx.x * 8 + wave) * 16;
    const int m       = lane & 15;      // row within the 16-row tile
    const int half    = lane >> 4;      // 0: K=0,1   1: K=2,3 (f32 A-matrix layout)

    const int row = rowbase + m;
    const int lbl = labels[row];        // both wave halves read the same 16 labels (cached)

    const float* __restrict__ frow = feat   + (size_t)row * FD;
    const float* __restrict__ crow = center + (size_t)lbl * FD;

    // B matrix (4x16 f32): ones in column N=0 only -> D[m][0] = sum_k A[m][k].
    v2f bmat;
    bmat.x = (m == 0) ? 1.0f : 0.0f;
    bmat.y = bmat.x;

    v8f acc = {};   // 16x16 f32 accumulator, we use column 0 only

    const int cbase = 2 * half;
#pragma unroll 4
    for (int k = 0; k < FD / 4; ++k) {
        const int c0 = k * 4 + cbase;
        v2f f = *(const v2f*)(frow + c0);
        v2f c = *(const v2f*)(crow + c0);
        v2f d;  d.x = f.x - c.x;   d.y = f.y - c.y;
        v2f a;  a.x = d.x * d.x;   a.y = d.y * d.y;
        // D = A(16x4 f32) * B(4x16 f32) + C ; exact f32, RNE
        acc = __builtin_amdgcn_wmma_f32_16x16x4_f32(
                  false, a, false, bmat, (short)0, acc, false, false);
    }

    // Column 0 of D: lane 0 holds M=0..7 (VGPR 0..7), lane 16 holds M=8..15.
    float part = 0.0f;
#pragma unroll
    for (int i = 0; i < 8; ++i) {
        float s    = acc[i];
        float dist = fminf(fmaxf(sqrtf(s), 1e-12f), 1e12f);
        part += dist;
    }
    float tot = __shfl(part, 0, 32) + __shfl(part, 16, 32);
    if (lane == 0) atomicAdd(&ws[0], tot);

    // Tail: rows belonging to class 998/999 feed the inter-loss centroids.
    // lbl lives on lanes 0..15 (and mirrored on 16..31); broadcast per row.
    for (int r = 0; r < 16; ++r) {
        int lr = __shfl(lbl, r, 32);          // wave-uniform
        if (lr >= 998) {
            const int which = lr - 998;
            const float* __restrict__ fr = feat + (size_t)(rowbase + r) * FD;
            float* s = ws + WS_SUMS + which * FD;
            for (int j = lane; j < FD; j += 32) atomicAdd(&s[j], fr[j]);
            if (lane == 0) atomicAdd(&ws[1 + which], 1.0f);
        }
    }
}

// Single block of 512 threads: build the two centroids, reduce the squared
// distance, emit both outputs.
__global__ void __launch_bounds__(512)
center_loss_final_kernel(const float* __restrict__ center,
                         const float* __restrict__ ws,
                         float* __restrict__ out) {
    __shared__ float red[FD];
    const int j = threadIdx.x;          // 0..511

    const float c0 = fmaxf(ws[1], 1.0f);
    const float c1 = fmaxf(ws[2], 1.0f);
    const float a  = (center[(size_t)998 * FD + j] + ws[WS_SUMS + j])      / c0;
    const float b  = (center[(size_t)999 * FD + j] + ws[WS_SUMS + FD + j]) / c1;
    const float d  = a - b;
    red[j] = d * d;
    __syncthreads();

#pragma unroll
    for (int s = 256; s > 0; s >>= 1) {
        if (j < s) red[j] += red[j + s];
        __syncthreads();
    }

    if (j == 0) {
        const float dl = sqrtf(red[0]);
        out[0] = ws[0] * (1.0f / (float)NB);                    // intra_loss
        out[1] = (2.0f / dl) * (1.0f / (1000.0f * 999.0f));     // inter_loss
    }
}

extern "C" void kernel_launch(void* const* d_in, const int* in_sizes, int n_in,
                              void* d_out, int out_size, void* d_ws, size_t ws_size,
                              hipStream_t stream) {
    const float* feat   = (const float*)d_in[0];   // [32768, 512] f32
    const int*   labels = (const int*)  d_in[1];   // [32768]      i32
    const float* center = (const float*)d_in[2];   // [1000, 512]  f32
    float*       out    = (float*)d_out;           // [2] f32 (intra, inter)
    float*       ws     = (float*)d_ws;

    const int ws_elems = WS_SUMS + 2 * FD;         // 1040 floats
    zero_ws_kernel<<<(ws_elems + 255) / 256, 256, 0, stream>>>(ws, ws_elems);

    center_loss_main_kernel<<<NB / 128, 256, 0, stream>>>(feat, labels, center, ws);

    center_loss_final_kernel<<<1, FD, 0, stream>>>(center, ws, out);
}